// GlobalAttention_5016521801997
// MI455X (gfx1250) — compile-verified
//
#include <hip/hip_runtime.h>
#include <hip/hip_bf16.h>
#include <stdint.h>

#define SL 1024
#define BS 64
#define DS 1024
#define DT 1024
#define NEGVAL -9999.9f

typedef float v2f __attribute__((ext_vector_type(2)));
typedef float v4f __attribute__((ext_vector_type(4)));
typedef float v8f __attribute__((ext_vector_type(8)));

// ---------------------------------------------------------------------------
// Kernel 1: v[b,d] = sum_t h_t[b,t] * Wa[d,t]   (v = h_t @ Wa^T)
// Full-fp32 WMMA: V_WMMA_F32_16X16X4_F32, one wave per 16x16 tile of v.
//   A = h_t  (M=b, K=t), row-major stride DT
//   B = Wa^T (K=t, N=d)  -> element B[k][n] = Wa[n*DT + k]
// ---------------------------------------------------------------------------
__global__ void __launch_bounds__(32) v_gemm_kernel(const float* __restrict__ h_t,
                                                    const float* __restrict__ Wa,
                                                    float* __restrict__ v)
{
    const int n0   = blockIdx.x * 16;   // d tile
    const int m0   = blockIdx.y * 16;   // b tile
    const int lane = threadIdx.x;       // 0..31 (wave32)
    const int r    = lane & 15;         // row (A) / column (B) within tile
    const int kh   = (lane >> 4) << 1;  // 0 for low half-wave, 2 for high

    v8f acc = {};
    for (int k0 = 0; k0 < DT; k0 += 4) {
        const int kb = k0 + kh;
        v2f a = *(const v2f*)(h_t + (size_t)(m0 + r) * DT + kb); // A[m][kb..kb+1]
        v2f b = *(const v2f*)(Wa  + (size_t)(n0 + r) * DT + kb); // B[kb..kb+1][n]
        acc = __builtin_amdgcn_wmma_f32_16x16x4_f32(
            /*neg_a=*/false, a, /*neg_b=*/false, b,
            /*c_mod=*/(short)0, acc, /*reuse_a=*/false, /*reuse_b=*/false);
    }

    const int half = lane >> 4;
    const int n    = n0 + (lane & 15);
#pragma unroll
    for (int j = 0; j < 8; ++j) {
        const int m = m0 + j + 8 * half;
        v[(size_t)m * DS + n] = acc[j];
    }
}

__device__ __forceinline__ float waveReduceSum(float x) {
#pragma unroll
    for (int off = 16; off > 0; off >>= 1)
        x += __shfl_xor(x, off, 32);
    return x;
}

__device__ __forceinline__ float waveReduceMax(float x) {
#pragma unroll
    for (int off = 16; off > 0; off >>= 1)
        x = fmaxf(x, __shfl_xor(x, off, 32));
    return x;
}

// ---------------------------------------------------------------------------
// Kernel 2: scores[s,b] = mask[s,b] ? h_s[s,b,:] . v[b,:] : NEG
// One wave32 per (s,b) row; 8 rows per 256-thread block. Pure streaming +
// shuffle reduce: no LDS, no barriers. First pass over h_s (256 MB).
// ---------------------------------------------------------------------------
__global__ void __launch_bounds__(256) scores_kernel(const float* __restrict__ h_s,
                                                     const float* __restrict__ v,
                                                     const uint8_t* __restrict__ mask,
                                                     float* __restrict__ scores)
{
    const int row  = blockIdx.x * 8 + (threadIdx.x >> 5);  // sb = s*BS + b
    const int lane = threadIdx.x & 31;
    const int b    = row & (BS - 1);

    const float* hrow = h_s + (size_t)row * DS;
    const float* vrow = v + (size_t)b * DS;

    float p = 0.f;
#pragma unroll
    for (int i = 0; i < 8; ++i) {
        const int d = (lane + i * 32) * 4;               // 512B contiguous per wave op
        const v4f hv = *(const v4f*)(hrow + d);
        const v4f vv = *(const v4f*)(vrow + d);          // 4KB/b row, L2-resident
        p = fmaf(hv.x, vv.x, p);
        p = fmaf(hv.y, vv.y, p);
        p = fmaf(hv.z, vv.z, p);
        p = fmaf(hv.w, vv.w, p);
    }
    p = waveReduceSum(p);
    if (lane == 0) scores[row] = mask[row] ? p : NEGVAL;
}

// ---------------------------------------------------------------------------
// Kernel 3: softmax over s (axis 0) per batch column b.
// Writes almt (internal, [s,b]) and out_almtT ([b,s], first output), and
// zeroes the ctx output region so kernel 4 can atomically accumulate.
// ---------------------------------------------------------------------------
__global__ void __launch_bounds__(256) softmax_kernel(const float* __restrict__ scores,
                                                      float* __restrict__ almt,
                                                      float* __restrict__ out) // [almtT | ctx]
{
    const int b   = blockIdx.x;
    const int tid = threadIdx.x;
    __shared__ float red[8];
    __shared__ float bcast;
    const int wave = tid >> 5;
    const int lane = tid & 31;

    float x[4];
#pragma unroll
    for (int i = 0; i < 4; ++i) x[i] = scores[(tid + i * 256) * BS + b];

    // block max
    float m = fmaxf(fmaxf(x[0], x[1]), fmaxf(x[2], x[3]));
    m = waveReduceMax(m);
    if (lane == 0) red[wave] = m;
    __syncthreads();
    if (tid == 0) {
        float t = red[0];
#pragma unroll
        for (int i = 1; i < 8; ++i) t = fmaxf(t, red[i]);
        bcast = t;
    }
    __syncthreads();
    m = bcast;

    // block sum of exp
    float e[4];
    float s = 0.f;
#pragma unroll
    for (int i = 0; i < 4; ++i) { e[i] = __expf(x[i] - m); s += e[i]; }
    s = waveReduceSum(s);
    __syncthreads();
    if (lane == 0) red[wave] = s;
    __syncthreads();
    if (tid == 0) {
        float t = 0.f;
#pragma unroll
        for (int i = 0; i < 8; ++i) t += red[i];
        bcast = t;
    }
    __syncthreads();
    const float inv = 1.0f / bcast;

#pragma unroll
    for (int i = 0; i < 4; ++i) {
        const int sidx = tid + i * 256;
        const float a  = e[i] * inv;
        almt[sidx * BS + b]        = a;  // internal [s,b] for pooling pass
        out[(size_t)b * SL + sidx] = a;  // output 0: almt_distr.T, (BS, SL)
    }

    // zero ctx output region (output 1) for atomic accumulation
    float* ctx = out + (size_t)BS * SL;
    v4f z = {0.f, 0.f, 0.f, 0.f};
    *(v4f*)(ctx + (size_t)b * DS + tid * 4) = z;
}

// ---------------------------------------------------------------------------
// Kernel 4: ctx[b,d] = sum_s almt[s,b] * h_s[s,b,d]
// Second (final) pass over h_s: nontemporal loads (last-use; 256MB > 192MB L2)
// and reverse s-chunk order so the tail of pass 1 — the freshest L2 lines —
// is consumed first. Register accumulation over 64 s values, then 4 atomics.
// ---------------------------------------------------------------------------
__global__ void __launch_bounds__(256) ctx_kernel(const float* __restrict__ h_s,
                                                  const float* __restrict__ almt,
                                                  float* __restrict__ ctx)
{
    const int b  = blockIdx.x;
    const int s0 = (SL / 64 - 1 - blockIdx.y) * 64;  // reverse-order streaming
    const int d  = threadIdx.x * 4;

    float ax = 0.f, ay = 0.f, az = 0.f, aw = 0.f;
    for (int s = s0; s < s0 + 64; ++s) {
        const float w  = almt[s * BS + b];
        const v4f   hv = __builtin_nontemporal_load(
            (const v4f*)(h_s + ((size_t)(s * BS + b)) * DS + d));
        ax = fmaf(w, hv.x, ax);
        ay = fmaf(w, hv.y, ay);
        az = fmaf(w, hv.z, az);
        aw = fmaf(w, hv.w, aw);
    }
    float* p = ctx + (size_t)b * DS + d;
    atomicAdd(p + 0, ax);
    atomicAdd(p + 1, ay);
    atomicAdd(p + 2, az);
    atomicAdd(p + 3, aw);
}

extern "C" void kernel_launch(void* const* d_in, const int* in_sizes, int n_in,
                              void* d_out, int out_size, void* d_ws, size_t ws_size,
                              hipStream_t stream) {
    const float*   h_t  = (const float*)d_in[0];
    const float*   h_s  = (const float*)d_in[1];
    const uint8_t* mask = (const uint8_t*)d_in[2];
    const float*   Wa   = (const float*)d_in[3];

    float* out = (float*)d_out;          // [ almt_T (BS*SL) | ctx (BS*DS) ]
    float* ws  = (float*)d_ws;
    float* v      = ws;                  // BS*DS   = 65536 floats
    float* scores = v + BS * DS;         // SL*BS   = 65536 floats
    float* almt   = scores + SL * BS;    // SL*BS   = 65536 floats  (768 KB total)

    // v = h_t @ Wa^T via fp32 WMMA (one wave per 16x16 tile)
    v_gemm_kernel<<<dim3(DS / 16, BS / 16), 32, 0, stream>>>(h_t, Wa, v);
    // pass 1 over h_s: masked scores (wave-per-row streaming)
    scores_kernel<<<(SL * BS) / 8, 256, 0, stream>>>(h_s, v, mask, scores);
    // per-batch softmax over the sequence axis; also zeroes ctx output
    softmax_kernel<<<BS, 256, 0, stream>>>(scores, almt, out);
    // pass 2 over h_s: weighted pooling (nontemporal, reverse order)
    ctx_kernel<<<dim3(BS, SL / 64), 256, 0, stream>>>(h_s, almt, out + (size_t)BS * SL);
}